// GConv_28441273434764
// MI455X (gfx1250) — compile-verified
//
#include <hip/hip_runtime.h>

// ---- problem constants (match reference) ----
constexpr int N_NODES = 50000;
constexpr int N_EDGES = 800000;
constexpr int IN_DIM  = 256;
constexpr int HID     = 128;

typedef __attribute__((ext_vector_type(2))) float v2f;
typedef __attribute__((ext_vector_type(8))) float v8f;

// =====================================================================
// GEMM: OUT[N, HID] = X[N, K] @ W[HID, K]^T  using V_WMMA_F32_16X16X4_F32
// One block = 256 threads = 8 waves. Block handles a 16-row tile of X,
// each wave produces one 16x16 output tile (8 waves cover all 128 cols).
// X tile staged in LDS (row-padded by 4 floats -> conflict-free reads).
// =====================================================================
template<int K>
__global__ void __launch_bounds__(256)
gcn_gemm_wmma(const float* __restrict__ X, const float* __restrict__ W,
              float* __restrict__ OUT)
{
    constexpr int KP = K + 4;               // pad keeps 16B align, kills bank conflicts
    __shared__ float tileX[16 * KP];

    const int tid  = threadIdx.x;
    const int row0 = blockIdx.x * 16;

    // cooperative load: 16 rows x K floats (contiguous in memory) -> padded LDS
    const float4* Xv = reinterpret_cast<const float4*>(X + (size_t)row0 * K);
    constexpr int NV = 16 * K / 4;
    #pragma unroll
    for (int i = tid; i < NV; i += 256) {
        const int r = (i * 4) / K;
        const int c = (i * 4) % K;
        *reinterpret_cast<float4*>(&tileX[r * KP + c]) = Xv[i];
    }
    __syncthreads();

    const int wave = tid >> 5;
    const int lane = tid & 31;
    const int n0   = wave * 16;             // output-column tile for this wave
    const int mn   = lane & 15;             // A row / B column index
    const int koff = (lane >> 4) * 2;       // ISA: lanes 16..31 carry K+2,K+3

    const float* arow = &tileX[mn * KP];              // A: X row (from LDS)
    const float* wrow = W + (size_t)(n0 + mn) * K;    // B: W row n (B = W^T)

    v8f acc = {};
    #pragma unroll 8
    for (int k = 0; k < K; k += 4) {
        const int kb = k + koff;
        v2f a = *reinterpret_cast<const v2f*>(arow + kb);   // K=kb, kb+1
        v2f b = *reinterpret_cast<const v2f*>(wrow + kb);
        acc = __builtin_amdgcn_wmma_f32_16x16x4_f32(
                  /*neg_a=*/false, a, /*neg_b=*/false, b,
                  /*c_mod=*/(short)0, acc, /*reuse_a=*/false, /*reuse_b=*/false);
    }

    // C/D layout: VGPR i -> M = i (lanes 0-15), M = i+8 (lanes 16-31); N = lane&15
    const int colOut = n0 + mn;
    const int rbase  = row0 + 8 * (lane >> 4);
    #pragma unroll
    for (int i = 0; i < 8; ++i) {
        OUT[(size_t)(rbase + i) * HID + colOut] = acc[i];
    }
}

// =====================================================================
// out[i] = b[i % HID]   (seed accumulator with bias before scatter)
// =====================================================================
__global__ void __launch_bounds__(256)
gcn_init_bias(float* __restrict__ p, const float* __restrict__ b)
{
    const int i = blockIdx.x * 256 + threadIdx.x;   // grid sized exactly
    p[i] = b[i & (HID - 1)];
}

// =====================================================================
// hardware no-return fp32 atomic add (device scope -> L2 atomic units).
// Forces global_atomic_add_f32 (STOREcnt path) instead of a CAS loop.
// =====================================================================
__device__ __forceinline__ void hw_atomic_add_f32(float* p, float v)
{
    asm volatile("global_atomic_add_f32 %0, %1, off scope:SCOPE_DEV"
                 :: "v"(p), "v"(v)
                 : "memory");
}

// =====================================================================
// scatter: out[dst[e]] += w[e] * fts[src[e]]  -- one wave per edge,
// float4 per lane (wave covers all 128 feats), hw f32 atomics into L2.
// =====================================================================
__global__ void __launch_bounds__(256)
gcn_scatter(const float* __restrict__ fts, const int* __restrict__ src,
            const int* __restrict__ dst, const float* __restrict__ ew,
            float* __restrict__ out)
{
    const int e    = blockIdx.x * 8 + (threadIdx.x >> 5);  // 8 waves / block
    const int lane = threadIdx.x & 31;

    const int   s = src[e];
    const int   d = dst[e];
    const float w = ew[e];

    const float4 v = reinterpret_cast<const float4*>(fts + (size_t)s * HID)[lane];
    float* o = out + (size_t)d * HID + lane * 4;

    hw_atomic_add_f32(o + 0, w * v.x);
    hw_atomic_add_f32(o + 1, w * v.y);
    hw_atomic_add_f32(o + 2, w * v.z);
    hw_atomic_add_f32(o + 3, w * v.w);
}

// =====================================================================
// in-place PReLU with shared scalar alpha
// =====================================================================
__global__ void __launch_bounds__(256)
gcn_prelu(float* __restrict__ p, const float* __restrict__ a)
{
    const int i = blockIdx.x * 256 + threadIdx.x;   // grid sized exactly
    const float al = *a;
    const float v  = p[i];
    p[i] = (v >= 0.0f) ? v : al * v;
}

// =====================================================================
// host launcher
// =====================================================================
extern "C" void kernel_launch(void* const* d_in, const int* in_sizes, int n_in,
                              void* d_out, int out_size, void* d_ws, size_t ws_size,
                              hipStream_t stream)
{
    const float* x   = (const float*)d_in[0];
    const int*   ei  = (const int*)  d_in[1];
    const float* ew  = (const float*)d_in[2];
    const float* W1  = (const float*)d_in[3];
    const float* b1  = (const float*)d_in[4];
    const float* a1  = (const float*)d_in[5];
    const float* W2  = (const float*)d_in[6];
    const float* b2  = (const float*)d_in[7];
    const float* a2  = (const float*)d_in[8];
    float*       out = (float*)d_out;

    const int* src = ei;            // edge_index[0]
    const int* dst = ei + N_EDGES;  // edge_index[1]

    float* buf0 = (float*)d_ws;                      // fts   (25.6 MB)
    float* buf1 = buf0 + (size_t)N_NODES * HID;      // agg/z (25.6 MB)

    const dim3 blk(256);
    const int  gridRows = N_NODES / 16;              // 3125 (exact)
    const int  gridElem = (N_NODES * HID) / 256;     // 25000 (exact)
    const int  gridEdge = N_EDGES / 8;               // 100000 (exact)

    // ----- layer 1 -----
    gcn_gemm_wmma<IN_DIM><<<gridRows, blk, 0, stream>>>(x, W1, buf0);
    gcn_init_bias<<<gridElem, blk, 0, stream>>>(buf1, b1);
    gcn_scatter<<<gridEdge, blk, 0, stream>>>(buf0, src, dst, ew, buf1);
    gcn_prelu<<<gridElem, blk, 0, stream>>>(buf1, a1);

    // ----- layer 2 ----- (reuse buf0 for fts2; accumulate straight into d_out)
    gcn_gemm_wmma<HID><<<gridRows, blk, 0, stream>>>(buf1, W2, buf0);
    gcn_init_bias<<<gridElem, blk, 0, stream>>>(out, b2);
    gcn_scatter<<<gridEdge, blk, 0, stream>>>(buf0, src, dst, ew, out);
    gcn_prelu<<<gridElem, blk, 0, stream>>>(out, a2);
}